// MambaWrapper_36447092473918
// MI455X (gfx1250) — compile-verified
//
#include <hip/hip_runtime.h>
#include <hip/hip_bf16.h>

// ---------------------------------------------------------------------------
// Bidirectional Mamba block forward for MI455X (gfx1250), fp32 throughout.
// GEMMs run on the matrix pipe via V_WMMA_F32_16X16X4_F32 (wave32 WMMA).
// Each wave register-blocks MTxNT 16x16 tiles so one K=4 step issues
// (MT+NT) b64 loads against MT*NT WMMAs -> high flops/byte at the VMEM port.
// __launch_bounds__(256, 1) gives the compiler the full 256-VGPR budget so
// the 4x4 accumulator block (128 VGPRs) stays resident -- no scratch spills.
// ---------------------------------------------------------------------------

#define D_MODEL 768
#define D_INNER 1536
#define D_STATE 16
#define D_CONV  4
#define DT_RANK 48
#define B_SZ    2
#define SEQ_L   2048
#define M_ROWS  (B_SZ * SEQ_L)          // 4096
#define DBL_W   (DT_RANK + 2 * D_STATE) // 80

typedef __attribute__((ext_vector_type(2))) float v2f;
typedef __attribute__((ext_vector_type(8))) float v8f;

// ---------------------------------------------------------------------------
// WMMA register-blocked inner product: acc[MT][NT] += A-tiles x B-tiles.
// Fragment layouts per CDNA5 ISA (CDNA5 ISA):
//   A (16x4, MxK): lanes 0-15/16-31 both hold M=0..15; VGPR pair covers
//                  K {0,1} (lanes 0-15) / K {2,3} (lanes 16-31).
//   B (4x16, KxN): lane%16 indexes N, same K split by lane half.
// `arow[i]`/`brow[j]` are per-lane row pointers already offset by 2*laneHalf,
// so each lane streams one float2 per fragment per K-step.
// ---------------------------------------------------------------------------
template <int MT, int NT>
__device__ __forceinline__ void wmma_block(const float* (&arow)[MT],
                                           const float* (&brow)[NT],
                                           int K, v8f (&acc)[MT][NT]) {
#pragma unroll 2
  for (int kk = 0; kk < K; kk += 4) {
    v2f a[MT], b[NT];
#pragma unroll
    for (int i = 0; i < MT; ++i) a[i] = *(const v2f*)(arow[i] + kk);
#pragma unroll
    for (int j = 0; j < NT; ++j) b[j] = *(const v2f*)(brow[j] + kk);
#pragma unroll
    for (int i = 0; i < MT; ++i)
#pragma unroll
      for (int j = 0; j < NT; ++j)
        acc[i][j] = __builtin_amdgcn_wmma_f32_16x16x4_f32(
            /*neg_a=*/false, a[i], /*neg_b=*/false, b[j],
            /*c_mod=*/(short)0, acc[i][j], /*reuse_a=*/false, /*reuse_b=*/false);
  }
}

// C/D layout (32-bit 16x16): VGPR r, lanes 0-15 -> (M=r, N=lane),
//                                     lanes 16-31 -> (M=r+8, N=lane-16).

// ---------------------------------------------------------------------------
// Kernel A: xz = h @ in_proj^T  (M=4096, K=768, N=3072), 64x64 per wave.
// Columns n < D_INNER -> xraw, else -> zbuf (wave-uniform per 16-wide tile).
// Direction handled by reading h at reversed l for rev=1; all downstream
// buffers are in direction-local time.
// ---------------------------------------------------------------------------
__global__ void __launch_bounds__(256, 1)
gemm_inproj(const float* __restrict__ h,
            const float* __restrict__ W,   // [3072, 768]
            float* __restrict__ xraw,      // [M, D_INNER]
            float* __restrict__ zbuf,      // [M, D_INNER]
            int rev) {
  const int lane = threadIdx.x & 31;
  const int wave = threadIdx.x >> 5;
  const int half = lane >> 4;
  const int lr   = lane & 15;

  const int NG = (2 * D_INNER) / 64;                 // 48 column groups
  const int tile = blockIdx.x * 8 + wave;
  const int m0 = (tile / NG) * 64, n0 = (tile % NG) * 64;

  const float* arow[4];
#pragma unroll
  for (int mt = 0; mt < 4; ++mt) {
    const int m = m0 + mt * 16 + lr;
    const int b = m >> 11, l = m & (SEQ_L - 1);
    const int lsrc = rev ? (SEQ_L - 1 - l) : l;
    arow[mt] = h + (long)(b * SEQ_L + lsrc) * D_MODEL + 2 * half;
  }
  const float* brow[4];
#pragma unroll
  for (int nt = 0; nt < 4; ++nt)
    brow[nt] = W + (long)(n0 + nt * 16 + lr) * D_MODEL + 2 * half;

  v8f acc[4][4] = {};
  wmma_block<4, 4>(arow, brow, D_MODEL, acc);

#pragma unroll
  for (int nt = 0; nt < 4; ++nt) {
    const int nn0 = n0 + nt * 16;                    // wave-uniform
    float* dst;
    int nc;
    if (nn0 < D_INNER) { dst = xraw; nc = nn0 + lr; }
    else               { dst = zbuf; nc = nn0 - D_INNER + lr; }
#pragma unroll
    for (int mt = 0; mt < 4; ++mt)
#pragma unroll
      for (int r = 0; r < 8; ++r) {
        const int mm = m0 + mt * 16 + r + 8 * half;
        dst[(long)mm * D_INNER + nc] = acc[mt][nt][r];
      }
  }
}

// ---------------------------------------------------------------------------
// Kernel B: causal depthwise conv (width 4) + bias + SiLU, direction-local.
// ---------------------------------------------------------------------------
__global__ void conv_silu(const float* __restrict__ xraw,
                          const float* __restrict__ cw,   // [D_INNER, 4]
                          const float* __restrict__ cb,   // [D_INNER]
                          float* __restrict__ xconv) {
  const int idx = blockIdx.x * blockDim.x + threadIdx.x;
  if (idx >= M_ROWS * D_INNER) return;
  const int d = idx % D_INNER;
  const int bl = idx / D_INNER;
  const int l = bl & (SEQ_L - 1);
  const int b = bl >> 11;
  float acc = cb[d];
#pragma unroll
  for (int j = 0; j < D_CONV; ++j) {
    const int ls = l - (D_CONV - 1) + j;
    if (ls >= 0)
      acc += cw[d * D_CONV + j] * xraw[(long)(b * SEQ_L + ls) * D_INNER + d];
  }
  xconv[idx] = acc / (1.0f + __expf(-acc));   // silu
}

// ---------------------------------------------------------------------------
// Kernel C: dbl = x @ xproj^T  (M=4096, K=1536, N=80), 16x80 per wave.
// ---------------------------------------------------------------------------
__global__ void __launch_bounds__(256, 1)
gemm_xproj(const float* __restrict__ x,      // [M, D_INNER]
           const float* __restrict__ W,      // [80, D_INNER]
           float* __restrict__ dbl) {        // [M, 80]
  const int lane = threadIdx.x & 31;
  const int wave = threadIdx.x >> 5;
  const int half = lane >> 4;
  const int lr   = lane & 15;

  const int tile = blockIdx.x * 8 + wave;            // m-tile 0..255
  const int m0 = tile * 16;

  const float* arow[1] = { x + (long)(m0 + lr) * D_INNER + 2 * half };
  const float* brow[5];
#pragma unroll
  for (int nt = 0; nt < 5; ++nt)
    brow[nt] = W + (long)(nt * 16 + lr) * D_INNER + 2 * half;

  v8f acc[1][5] = {};
  wmma_block<1, 5>(arow, brow, D_INNER, acc);

#pragma unroll
  for (int nt = 0; nt < 5; ++nt)
#pragma unroll
    for (int r = 0; r < 8; ++r) {
      const int mm = m0 + r + 8 * half;
      dbl[(long)mm * DBL_W + nt * 16 + lr] = acc[0][nt][r];
    }
}

// ---------------------------------------------------------------------------
// Kernel D: dt = softplus(dbl[:, :48] @ dtproj_w^T + dtproj_b)
//           (M=4096, K=48, N=1536), 32x32 per wave.
// ---------------------------------------------------------------------------
__global__ void __launch_bounds__(256, 1)
gemm_dtproj(const float* __restrict__ dbl,   // [M, 80]
            const float* __restrict__ W,     // [D_INNER, 48]
            const float* __restrict__ bias,  // [D_INNER]
            float* __restrict__ dtbuf) {     // [M, D_INNER]
  const int lane = threadIdx.x & 31;
  const int wave = threadIdx.x >> 5;
  const int half = lane >> 4;
  const int lr   = lane & 15;

  const int NG = D_INNER / 32;                       // 48 column groups
  const int tile = blockIdx.x * 8 + wave;
  const int m0 = (tile / NG) * 32, n0 = (tile % NG) * 32;

  const float* arow[2];
  const float* brow[2];
#pragma unroll
  for (int t = 0; t < 2; ++t) {
    arow[t] = dbl + (long)(m0 + t * 16 + lr) * DBL_W + 2 * half;     // dt cols
    brow[t] = W + (long)(n0 + t * 16 + lr) * DT_RANK + 2 * half;
  }

  v8f acc[2][2] = {};
  wmma_block<2, 2>(arow, brow, DT_RANK, acc);

#pragma unroll
  for (int mt = 0; mt < 2; ++mt)
#pragma unroll
    for (int nt = 0; nt < 2; ++nt)
#pragma unroll
      for (int r = 0; r < 8; ++r) {
        const int mm = m0 + mt * 16 + r + 8 * half;
        const int nn = n0 + nt * 16 + lr;
        float v = acc[mt][nt][r] + bias[nn];
        v = (v > 20.0f) ? v : log1pf(__expf(v));     // softplus
        dtbuf[(long)mm * D_INNER + nn] = v;
      }
}

// ---------------------------------------------------------------------------
// Kernel E: selective scan (sequential over L, parallel over B*D_INNER).
// Each thread owns one (b, d) channel and its 16 states in registers.
// B/C rows are lane-uniform -> scalar (SMEM) loads, broadcast from L2.
// y = (scan + x*Dskip) * silu(z), written over the xraw buffer (dead here).
// ---------------------------------------------------------------------------
__global__ void scan_kernel(const float* __restrict__ xconv,
                            const float* __restrict__ dtbuf,
                            const float* __restrict__ dbl,
                            const float* __restrict__ zbuf,
                            const float* __restrict__ A_log,  // [D_INNER, 16]
                            const float* __restrict__ Dskip,  // [D_INNER]
                            float* __restrict__ ybuf) {
  const int td = blockIdx.x * blockDim.x + threadIdx.x;
  if (td >= B_SZ * D_INNER) return;
  const int d = td % D_INNER;
  const int b = td / D_INNER;

  float Ad[D_STATE], hs[D_STATE];
#pragma unroll
  for (int n = 0; n < D_STATE; ++n) {
    Ad[n] = -__expf(A_log[d * D_STATE + n]);
    hs[n] = 0.0f;
  }
  const float dsk = Dskip[d];

  for (int t = 0; t < SEQ_L; ++t) {
    const long row = (long)(b * SEQ_L + t);
    const float dt = dtbuf[row * D_INNER + d];
    const float xv = xconv[row * D_INNER + d];
    const float* bc = dbl + row * DBL_W;   // B at +48, C at +64 (uniform)
    const float dtx = dt * xv;
    float y = 0.0f;
#pragma unroll
    for (int n = 0; n < D_STATE; ++n) {
      const float dA = __expf(dt * Ad[n]);
      hs[n] = hs[n] * dA + dtx * bc[DT_RANK + n];
      y += hs[n] * bc[DT_RANK + D_STATE + n];
    }
    y += xv * dsk;
    const float z = zbuf[row * D_INNER + d];
    ybuf[row * D_INNER + d] = y * (z / (1.0f + __expf(-z)));
  }
}

// ---------------------------------------------------------------------------
// Kernel F: out (+)= y @ out_proj^T  (M=4096, K=1536, N=768), 32x64 per wave.
// dir 0 stores (initializes), dir 1 accumulates; l re-reversed for rev=1.
// ---------------------------------------------------------------------------
__global__ void __launch_bounds__(256, 1)
gemm_outproj(const float* __restrict__ y,     // [M, D_INNER]
             const float* __restrict__ W,     // [768, D_INNER]
             float* __restrict__ out,         // [B, L, 768]
             int rev) {
  const int lane = threadIdx.x & 31;
  const int wave = threadIdx.x >> 5;
  const int half = lane >> 4;
  const int lr   = lane & 15;

  const int NG = D_MODEL / 64;                       // 12 column groups
  const int tile = blockIdx.x * 8 + wave;
  const int m0 = (tile / NG) * 32, n0 = (tile % NG) * 64;

  const float* arow[2];
#pragma unroll
  for (int mt = 0; mt < 2; ++mt)
    arow[mt] = y + (long)(m0 + mt * 16 + lr) * D_INNER + 2 * half;
  const float* brow[4];
#pragma unroll
  for (int nt = 0; nt < 4; ++nt)
    brow[nt] = W + (long)(n0 + nt * 16 + lr) * D_INNER + 2 * half;

  v8f acc[2][4] = {};
  wmma_block<2, 4>(arow, brow, D_INNER, acc);

#pragma unroll
  for (int mt = 0; mt < 2; ++mt)
#pragma unroll
    for (int r = 0; r < 8; ++r) {
      const int mm = m0 + mt * 16 + r + 8 * half;
      const int bb = mm >> 11, ll = mm & (SEQ_L - 1);
      const int lout = rev ? (SEQ_L - 1 - ll) : ll;
      const long base = (long)(bb * SEQ_L + lout) * D_MODEL;
#pragma unroll
      for (int nt = 0; nt < 4; ++nt) {
        const long idx = base + n0 + nt * 16 + lr;
        if (rev) out[idx] += acc[mt][nt][r];
        else     out[idx]  = acc[mt][nt][r];
      }
    }
}

// ---------------------------------------------------------------------------
// Host launch. Workspace layout (floats), reused across both directions:
//   [0]            xraw / ybuf   (6291456)   -- scan overwrites dead xraw
//   [1*6291456]    zbuf          (6291456)
//   [2*6291456]    xconv         (6291456)
//   [3*6291456]    dtbuf         (6291456)
//   [4*6291456]    dbl           (327680)
// Total ~97.3 MB.
// ---------------------------------------------------------------------------
extern "C" void kernel_launch(void* const* d_in, const int* in_sizes, int n_in,
                              void* d_out, int out_size, void* d_ws, size_t ws_size,
                              hipStream_t stream) {
  const float* h = (const float*)d_in[0];
  float* out = (float*)d_out;

  const long CH = (long)M_ROWS * D_INNER;   // 6291456
  float* ws    = (float*)d_ws;
  float* xraw  = ws;            // doubles as ybuf
  float* zbuf  = ws + CH;
  float* xconv = ws + 2 * CH;
  float* dtbuf = ws + 3 * CH;
  float* dbl   = ws + 4 * CH;

  const int waves_in  = (M_ROWS / 64) * ((2 * D_INNER) / 64);  // 3072
  const int waves_xp  = (M_ROWS / 16);                         // 256
  const int waves_dt  = (M_ROWS / 32) * (D_INNER / 32);        // 6144
  const int waves_out = (M_ROWS / 32) * (D_MODEL / 64);        // 1536

  for (int dir = 0; dir < 2; ++dir) {
    const float* in_proj  = (const float*)d_in[1 + dir * 9 + 0];
    const float* conv_w   = (const float*)d_in[1 + dir * 9 + 1];
    const float* conv_b   = (const float*)d_in[1 + dir * 9 + 2];
    const float* xproj    = (const float*)d_in[1 + dir * 9 + 3];
    const float* dtproj_w = (const float*)d_in[1 + dir * 9 + 4];
    const float* dtproj_b = (const float*)d_in[1 + dir * 9 + 5];
    const float* A_log    = (const float*)d_in[1 + dir * 9 + 6];
    const float* Dskip    = (const float*)d_in[1 + dir * 9 + 7];
    const float* out_proj = (const float*)d_in[1 + dir * 9 + 8];

    gemm_inproj<<<waves_in / 8, 256, 0, stream>>>(h, in_proj, xraw, zbuf, dir);
    conv_silu<<<(int)((CH + 255) / 256), 256, 0, stream>>>(xraw, conv_w, conv_b, xconv);
    gemm_xproj<<<waves_xp / 8, 256, 0, stream>>>(xconv, xproj, dbl);
    gemm_dtproj<<<waves_dt / 8, 256, 0, stream>>>(dbl, dtproj_w, dtproj_b, dtbuf);
    scan_kernel<<<(B_SZ * D_INNER + 255) / 256, 256, 0, stream>>>(
        xconv, dtbuf, dbl, zbuf, A_log, Dskip, xraw /*ybuf*/);
    gemm_outproj<<<waves_out / 8, 256, 0, stream>>>(xraw, out_proj, out, dir);
  }
}